// ImbalancedFocalLoss_52819507806486
// MI455X (gfx1250) — compile-verified
//
#include <hip/hip_runtime.h>
#include <hip/hip_bf16.h>
#include <stdint.h>

// ---------------------------------------------------------------------------
// ImbalancedFocalLoss on MI455X (gfx1250)
// Memory-bound streaming reduction: 288 MB in -> 3 floats out (~12.4 us @ 23.3 TB/s).
// Data movement uses the CDNA5 Tensor Data Mover (tensor_load_to_lds) with a
// per-wave double-buffered LDS pipeline synchronized via s_wait_tensorcnt.
// Toolchain note: this clang exposes the 6-arg tensor_load_to_lds builtin
// (uint32x4, int32x8, int32x4, int32x4, int32x8, i32 cpol).
// ---------------------------------------------------------------------------

#define B_SAMPLES   4194304
#define NCLS        8
#define NB          1024          // blocks in main grid (fixed -> deterministic)
#define BLOCK       256           // 8 waves (wave32)
#define WPB         8             // waves per block
#define S_TILE      128           // samples per wave-tile per task
#define TILE_FLOATS (S_TILE * NCLS)     // 1024 floats
#define TILE_BYTES  (TILE_FLOATS * 4)   // 4 KB
#define WAVE_LDS    (2 * 2 * TILE_BYTES)  // 2 bufs x (dir+vol) = 16 KB / wave
#define SMEM_MAIN   (WPB * WAVE_LDS)      // 128 KB
#define RED_OFF     SMEM_MAIN
#define SMEM_BYTES  (SMEM_MAIN + 2 * WPB * 4 + 64)

typedef unsigned int u32;
typedef unsigned long long u64;
typedef u32 uint4v __attribute__((ext_vector_type(4)));
typedef int  int8v __attribute__((ext_vector_type(8)));
typedef int  int4v __attribute__((ext_vector_type(4)));

#ifndef __has_builtin
#define __has_builtin(x) 0
#endif

#if __has_builtin(__builtin_amdgcn_tensor_load_to_lds) && \
    __has_builtin(__builtin_amdgcn_s_wait_tensorcnt)
#define USE_TDM 1
#else
#define USE_TDM 0
#endif

#if USE_TDM
// Issue one TDM load of a contiguous 4 KB tile (1024 f32) into LDS.
// D# layout per CDNA5 ISA ch.8: group0 = {count/flags, lds_addr, global_addr, type},
// group1 = {data_size, tensor dims, tile dims, strides}. 1-D tile: tile_dim0=1024,
// tensor_dim0=1024 (coords relative to tile start -> never OOB), stride=1024.
__device__ __forceinline__ void tdm_tile_load(u32 lds_off, const float* gptr) {
  u64 ga = (u64)(uintptr_t)gptr;
  uint4v g0;
  g0[0] = 1u;                                         // count=1, is_restore=0, gather off
  g0[1] = lds_off;                                    // lds_addr (bytes)
  g0[2] = (u32)ga;                                    // global_addr[31:0]
  g0[3] = ((u32)(ga >> 32) & 0x01FFFFFFu)             // global_addr[56:32]
        | 0x80000000u;                                // type=2 ("image")
  int8v g1;
  g1[0] = 0x00020000;                                 // data_size=2 (4B), wg_mask=0
  g1[1] = (int)(1024u << 16);                         // tensor_dim0[15:0] @ bits[63:48]
  g1[2] = (int)(1u << 16);                            // tensor_dim0 hi=0, tensor_dim1=1
  g1[3] = (int)(1024u << 16);                         // tile_dim0 = 1024
  g1[4] = 0;                                          // tile_dim1=0, tile_dim2=0 (unused)
  g1[5] = 1024;                                       // tensor_dim0_stride[31:0]
  g1[6] = 0;
  g1[7] = 0;
  int4v gz4;
  gz4[0] = 0; gz4[1] = 0; gz4[2] = 0; gz4[3] = 0;     // groups 2/3 unused (<=2D)
  int8v gz8;
  gz8[0] = 0; gz8[1] = 0; gz8[2] = 0; gz8[3] = 0;     // extra group (clang-23 form),
  gz8[4] = 0; gz8[5] = 0; gz8[6] = 0; gz8[7] = 0;     // zero-filled
  __builtin_amdgcn_tensor_load_to_lds(g0, g1, gz4, gz4, gz8, 0);
}
#endif

// Per-sample focal loss, all f32 (matches reference math).
__device__ __forceinline__ float focal_sample(const float* __restrict__ row, int lbl) {
  float4 a = *reinterpret_cast<const float4*>(row);
  float4 b = *reinterpret_cast<const float4*>(row + 4);
  float m = fmaxf(fmaxf(fmaxf(a.x, a.y), fmaxf(a.z, a.w)),
                  fmaxf(fmaxf(b.x, b.y), fmaxf(b.z, b.w)));
  float e0 = __expf(a.x - m), e1 = __expf(a.y - m);
  float e2 = __expf(a.z - m), e3 = __expf(a.w - m);
  float e4 = __expf(b.x - m), e5 = __expf(b.y - m);
  float e6 = __expf(b.z - m), e7 = __expf(b.w - m);
  float s = ((e0 + e1) + (e2 + e3)) + ((e4 + e5) + (e6 + e7));
  float el = e0;                          // branchless gather of exp(logit[label])
  el = (lbl == 1) ? e1 : el;  el = (lbl == 2) ? e2 : el;
  el = (lbl == 3) ? e3 : el;  el = (lbl == 4) ? e4 : el;
  el = (lbl == 5) ? e5 : el;  el = (lbl == 6) ? e6 : el;
  el = (lbl == 7) ? e7 : el;
  float pt = el / s;
  float alpha_t = (lbl == 0) ? 0.75f : 0.25f;         // 1-ALPHA : ALPHA
  float om = 1.0f - pt;
  float loss = -alpha_t * om * om * __logf(pt + 1e-8f);
  return (lbl == 1) ? loss * 1.5f : loss;             // class-1 boost
}

__global__ void __launch_bounds__(BLOCK) focal_main(
    const float* __restrict__ gDir, const int* __restrict__ lDir,
    const float* __restrict__ gVol, const int* __restrict__ lVol,
    float* __restrict__ partial)
{
  extern __shared__ __align__(16) unsigned char smem[];
  const int lane = threadIdx.x & 31;
  const int wid  = threadIdx.x >> 5;
  const int NT   = B_SAMPLES / S_TILE;    // 32768 wave-tiles
  const int TOTW = NB * WPB;              // total waves in grid
  int t = (int)blockIdx.x * WPB + wid;

  float pd = 0.0f, pv = 0.0f;

#if USE_TDM
  const u32 ldsBase = ((u32)(uintptr_t)(void*)smem) + (u32)(wid * WAVE_LDS);
  int buf = 0;
  if (t < NT) {   // prologue: prefetch first tile pair
    tdm_tile_load(ldsBase,              gDir + (u64)t * TILE_FLOATS);
    tdm_tile_load(ldsBase + TILE_BYTES, gVol + (u64)t * TILE_FLOATS);
  }
  while (t < NT) {
    const int tn = t + TOTW;
    if (tn < NT) {  // issue next stage, keep it in flight while computing
      u32 nb = ldsBase + (u32)((buf ^ 1) * (2 * TILE_BYTES));
      tdm_tile_load(nb,              gDir + (u64)tn * TILE_FLOATS);
      tdm_tile_load(nb + TILE_BYTES, gVol + (u64)tn * TILE_FLOATS);
      __builtin_prefetch(lDir + (u64)tn * S_TILE, 0, 1);
      __builtin_prefetch(lVol + (u64)tn * S_TILE, 0, 1);
      __builtin_amdgcn_s_wait_tensorcnt(2);   // 2 newest may remain outstanding
    } else {
      __builtin_amdgcn_s_wait_tensorcnt(0);   // drain for final tile
    }
    const float* dT = (const float*)(smem + wid * WAVE_LDS + buf * (2 * TILE_BYTES));
    const float* vT = dT + TILE_FLOATS;
    const int sbase = t * S_TILE;
#pragma unroll
    for (int i = 0; i < S_TILE / 32; ++i) {
      int j  = lane + 32 * i;
      int ld = lDir[sbase + j];
      int lv = lVol[sbase + j];
      pd += focal_sample(dT + j * NCLS, ld);
      pv += focal_sample(vT + j * NCLS, lv);
    }
    buf ^= 1;
    t = tn;
  }
#else
  // Fallback: direct coalesced global reads (no TDM builtin on this toolchain).
  for (; t < NT; t += TOTW) {
    const float* dT = gDir + (u64)t * TILE_FLOATS;
    const float* vT = gVol + (u64)t * TILE_FLOATS;
    const int sbase = t * S_TILE;
#pragma unroll
    for (int i = 0; i < S_TILE / 32; ++i) {
      int j  = lane + 32 * i;
      int ld = lDir[sbase + j];
      int lv = lVol[sbase + j];
      pd += focal_sample(dT + j * NCLS, ld);
      pv += focal_sample(vT + j * NCLS, lv);
    }
  }
#endif

  // wave32 butterfly reduction
  for (int o = 16; o > 0; o >>= 1) {
    pd += __shfl_down(pd, o, 32);
    pv += __shfl_down(pv, o, 32);
  }
  float* red = (float*)(smem + RED_OFF);
  if (lane == 0) { red[2 * wid] = pd; red[2 * wid + 1] = pv; }
  __syncthreads();
  if (threadIdx.x == 0) {
    float sd = 0.0f, sv = 0.0f;
#pragma unroll
    for (int w = 0; w < WPB; ++w) { sd += red[2 * w]; sv += red[2 * w + 1]; }
    partial[2 * blockIdx.x]     = sd;
    partial[2 * blockIdx.x + 1] = sv;
  }
}

__global__ void __launch_bounds__(256) focal_finalize(
    const float* __restrict__ partial, float* __restrict__ out)
{
  __shared__ float rd[8], rv[8];
  float sd = 0.0f, sv = 0.0f;
  for (int i = threadIdx.x; i < NB; i += 256) {
    sd += partial[2 * i];
    sv += partial[2 * i + 1];
  }
  for (int o = 16; o > 0; o >>= 1) {
    sd += __shfl_down(sd, o, 32);
    sv += __shfl_down(sv, o, 32);
  }
  const int lane = threadIdx.x & 31, wid = threadIdx.x >> 5;
  if (lane == 0) { rd[wid] = sd; rv[wid] = sv; }
  __syncthreads();
  if (threadIdx.x == 0) {
    float d = 0.0f, v = 0.0f;
#pragma unroll
    for (int w = 0; w < 8; ++w) { d += rd[w]; v += rv[w]; }
    const float inv = 1.0f / (float)B_SAMPLES;   // exact (power of two)
    float ldir = d * inv, lvol = v * inv;
    out[0] = ldir + 0.5f * lvol;   // total = 1.0*dir + 0.5*vol
    out[1] = ldir;
    out[2] = lvol;
  }
}

extern "C" void kernel_launch(void* const* d_in, const int* in_sizes, int n_in,
                              void* d_out, int out_size, void* d_ws, size_t ws_size,
                              hipStream_t stream) {
  (void)in_sizes; (void)n_in; (void)out_size; (void)ws_size;
  const float* gDir = (const float*)d_in[0];
  const int*   lDir = (const int*)d_in[1];
  const float* gVol = (const float*)d_in[2];
  const int*   lVol = (const int*)d_in[3];
  float* out     = (float*)d_out;
  float* partial = (float*)d_ws;   // NB*2 floats, fully rewritten every call

  (void)hipFuncSetAttribute((const void*)focal_main,
                            hipFuncAttributeMaxDynamicSharedMemorySize, SMEM_BYTES);
  focal_main<<<NB, BLOCK, SMEM_BYTES, stream>>>(gDir, lDir, gVol, lVol, partial);
  focal_finalize<<<1, 256, 0, stream>>>(partial, out);
}